// MPClayer_58411555226051
// MI455X (gfx1250) — compile-verified
//
#include <hip/hip_runtime.h>
#include <hip/hip_bf16.h>

// MPC layer, algebraically reduced.
// NH=254, NSTEP=32, N_STATE=256, NBATCH=256.

#define NHD   254
#define NSTP  32
#define NST   256          // N_STATE = NH + 2 = 256 (WMMA-friendly)
#define NB    256
#define DTC   (1.0f / 60.0f)
#define EPSR  1e-4f
#define PQC   5.0f
#define QVC   200.0f
#define QAC   1.0f

typedef float v2f __attribute__((ext_vector_type(2)));
typedef float v8f __attribute__((ext_vector_type(8)));

// ---------------------------------------------------------------------------
// Kernel 1: scalar/closed-form prep. One workgroup, everything in LDS.
//   X0   [256 x 256] : [x | gripper_p | own_gripper_v]
//   PmT  [32 x 256]  : (-p_final) transposed  -> B-fragment loads become b64
//   GT   [32 x 32]   : (Qinv @ M) transposed  -> B-fragment loads become b64
//   where M[j,k] = dt^2 (k - j + 0.5), j <= k
// ---------------------------------------------------------------------------
__global__ __launch_bounds__(256) void mpc_prep(
    const float* __restrict__ x,  const float* __restrict__ gp,
    const float* __restrict__ ov, const float* __restrict__ Af,
    const float* __restrict__ Lq,
    float* __restrict__ X0, float* __restrict__ PmT, float* __restrict__ GT)
{
    __shared__ float sAf[NHD];
    __shared__ float sz[NHD];
    __shared__ float sy[NST];
    __shared__ float red[256];
    __shared__ float Qf[NSTP * NSTP];
    __shared__ float Qi[NSTP * NSTP];
    __shared__ float colk[NSTP];
    __shared__ float S0[NSTP], S1[NSTP], S2[NSTP];
    __shared__ float cshared;

    const int t = threadIdx.x;

    // --- build X0 (thread t owns column t across all batch rows) ---
    for (int b = 0; b < NB; ++b) {
        float v;
        if (t < NHD)       v = x[b * NHD + t];
        else if (t == NHD) v = gp[b];
        else               v = ov[b];
        X0[b * NST + t] = v;
    }

    if (t < NHD) sAf[t] = Af[t];
    __syncthreads();

    // --- z = Lq^T Af  (coalesced: lane t reads column t of each row) ---
    if (t < NHD) {
        float acc = 0.0f;
        for (int i = 0; i < NHD; ++i) acc += Lq[i * NHD + t] * sAf[i];
        sz[t] = acc;
    }
    __syncthreads();

    // --- y = Lq z + eps Af   (Qtl @ Af without forming Qtl) ---
    if (t < NHD) {
        float acc = EPSR * sAf[t];
        const float* row = Lq + t * NHD;
        for (int j = 0; j < NHD; ++j) acc += row[j] * sz[j];
        sy[t] = acc;
    } else if (t < NST) {
        sy[t] = 0.0f;
    }
    __syncthreads();

    // --- c = Af . y  (tree reduction) ---
    red[t] = (t < NHD) ? sAf[t] * sy[t] : 0.0f;
    __syncthreads();
    for (int s = 128; s > 0; s >>= 1) {
        if (t < s) red[t] += red[t + s];
        __syncthreads();
    }
    if (t == 0) cshared = red[0];
    __syncthreads();
    const float c = cshared;

    // --- weighted prefix sums over steps: w_i = 1 (i<31), 5 (i==31) ---
    if (t < NSTP) {
        float s0 = 0.f, s1 = 0.f, s2 = 0.f;
        for (int i = t; i < NSTP; ++i) {
            float w = (i == NSTP - 1) ? PQC : 1.0f;
            s0 += w;
            s1 += w * (float)(i - t);
            s2 += w * (float)(i + 1) * (float)(i - t);
        }
        S0[t] = s0; S1[t] = s1; S2[t] = s2;
    }
    __syncthreads();

    // --- Q_final[a,b] = 2(Qa I + S^T Q S) + eps I (closed form) ---
    for (int e = t; e < NSTP * NSTP; e += 256) {
        int a = e >> 5, b = e & 31;
        int lo = (a > b) ? a : b;
        float acc = 0.0f;
        for (int i = lo; i < NSTP; ++i) {
            float w = (i == NSTP - 1) ? PQC : 1.0f;
            acc += w * ((float)(i - a) * (float)(i - b) * c + QVC);
        }
        float q = 2.0f * DTC * DTC * acc;
        if (a == b) q += 2.0f * QAC + EPSR;
        Qf[e] = q;
        Qi[e] = (a == b) ? 1.0f : 0.0f;
    }
    __syncthreads();

    // --- Gauss-Jordan inverse of SPD 32x32 (no pivoting needed) ---
    for (int k = 0; k < NSTP; ++k) {
        if (t < NSTP) colk[t] = Qf[t * NSTP + k];   // snapshot pivot column
        __syncthreads();
        float pinv = 1.0f / colk[k];
        if (t < NSTP)            Qf[k * NSTP + t] *= pinv;
        else if (t < 2 * NSTP)   Qi[k * NSTP + (t - NSTP)] *= pinv;
        __syncthreads();
        for (int e = t; e < NSTP * 2 * NSTP; e += 256) {
            int r = e >> 6, jj = e & 63;
            if (r != k) {
                float f = colk[r];
                if (jj < NSTP) Qf[r * NSTP + jj] -= f * Qf[k * NSTP + jj];
                else           Qi[r * NSTP + (jj - NSTP)] -= f * Qi[k * NSTP + (jj - NSTP)];
            }
        }
        __syncthreads();
    }

    // --- GT[n,k] = (Qinv @ M)[k,n] ;  M[j,n] = dt^2 (n - j + 0.5), j <= n ---
    for (int e = t; e < NSTP * NSTP; e += 256) {
        int n = e >> 5, k = e & 31;
        float acc = 0.0f;
        for (int j = 0; j <= n; ++j)
            acc += Qi[k * NSTP + j] * (DTC * DTC) * ((float)(n - j) + 0.5f);
        GT[n * NSTP + k] = acc;
    }

    // --- PmT[b,h] = -p_final[h,b]   [32 x 256], coalesced writes ---
    for (int e = t; e < NSTP * NST; e += 256) {
        int b = e >> 8, h = e & 255;
        float v;
        if (h < NHD)       v = -2.0f * DTC * S1[b] * sy[h];
        else if (h == NHD) v = 0.0f;
        else               v = -2.0f * DTC * (c * S2[b] + QVC * S0[b]);
        PmT[e] = v;   // e == b*NST + h
    }
}

// ---------------------------------------------------------------------------
// WMMA fragment layouts (V_WMMA_F32_16X16X4_F32, wave32):
//  A 16x4:  lane<16 -> M=lane, K={0,1}; lane>=16 -> M=lane-16, K={2,3}
//  B 4x16:  lane<16 -> N=lane, K={0,1}; lane>=16 -> N=lane-16, K={2,3}
//  C 16x16: VGPR r -> M = r + (lane>=16 ? 8 : 0), N = lane & 15
//  With B stored transposed (ldb = K-stride 1), both operands load as b64.
// ---------------------------------------------------------------------------

// Kernel 2: n_pxu = X0 @ (-p_final)   [256x256] x [256x32] -> [256x32]
__global__ __launch_bounds__(256) void mpc_gemm1(
    const float* __restrict__ X0, const float* __restrict__ PmT,
    float* __restrict__ PXU)
{
    const int wave = (blockIdx.x << 3) + (threadIdx.x >> 5);  // 0..31
    const int lane = threadIdx.x & 31;
    const int rt = wave >> 1;            // 16 row tiles
    const int ct = wave & 1;             // 2 col tiles
    const int m    = lane & 15;
    const int kOff = (lane >> 4) << 1;   // 0 or 2
    const int rowBase = rt << 4;
    const int colBase = ct << 4;

    v8f acc = {};
    const float* arow = X0  + (rowBase + m) * NST;
    const float* brow = PmT + (colBase + m) * NST;   // transposed: K contiguous
    for (int kc = 0; kc < NST / 4; ++kc) {
        const int kb = kc * 4 + kOff;                // even -> 8B aligned
        v2f a = *(const v2f*)(arow + kb);            // global_load_b64
        v2f b = *(const v2f*)(brow + kb);            // global_load_b64
        acc = __builtin_amdgcn_wmma_f32_16x16x4_f32(
            false, a, false, b, (short)0, acc, false, false);
    }

    const int mHi = (lane >> 4) << 3;    // 0 or 8
#pragma unroll
    for (int r = 0; r < 8; ++r) {
        int row = rowBase + r + mHi;
        PXU[row * NSTP + colBase + m] = acc[r];
    }
}

// Kernel 3: posi = PXU @ (Qinv M) + (gp[b] + (k+1) dt ov[b])  [256x32]x[32x32]
__global__ __launch_bounds__(256) void mpc_gemm2(
    const float* __restrict__ PXU, const float* __restrict__ GT,
    const float* __restrict__ gp,  const float* __restrict__ ov,
    float* __restrict__ out)
{
    const int wave = (blockIdx.x << 3) + (threadIdx.x >> 5);
    const int lane = threadIdx.x & 31;
    const int rt = wave >> 1;
    const int ct = wave & 1;
    const int m    = lane & 15;
    const int kOff = (lane >> 4) << 1;
    const int rowBase = rt << 4;
    const int colBase = ct << 4;

    v8f acc = {};
    const float* arow = PXU + (rowBase + m) * NSTP;
    const float* brow = GT  + (colBase + m) * NSTP;  // transposed: K contiguous
#pragma unroll
    for (int kc = 0; kc < NSTP / 4; ++kc) {
        const int kb = kc * 4 + kOff;
        v2f a = *(const v2f*)(arow + kb);
        v2f b = *(const v2f*)(brow + kb);
        acc = __builtin_amdgcn_wmma_f32_16x16x4_f32(
            false, a, false, b, (short)0, acc, false, false);
    }

    const int mHi  = (lane >> 4) << 3;
    const int kcol = colBase + m;
    const float kf = (float)(kcol + 1) * DTC;
#pragma unroll
    for (int r = 0; r < 8; ++r) {
        int row = rowBase + r + mHi;
        out[row * NSTP + kcol] = acc[r] + gp[row] + kf * ov[row];
    }
}

// ---------------------------------------------------------------------------
extern "C" void kernel_launch(void* const* d_in, const int* in_sizes, int n_in,
                              void* d_out, int out_size, void* d_ws, size_t ws_size,
                              hipStream_t stream) {
    const float* x  = (const float*)d_in[0];   // [256, 254]
    const float* gp = (const float*)d_in[1];   // [256]
    const float* ov = (const float*)d_in[2];   // [256]
    // d_in[3] = other_gripper_v : unused in the active forward path
    const float* Af = (const float*)d_in[4];   // [254, 1]
    const float* Lq = (const float*)d_in[5];   // [254, 254]

    float* ws  = (float*)d_ws;
    float* X0  = ws;                       // 65536 floats
    float* PmT = X0  + NB * NST;           //  8192 floats (32 x 256, transposed)
    float* GT  = PmT + NSTP * NST;         //  1024 floats (32 x 32, transposed)
    float* PXU = GT  + NSTP * NSTP;        //  8192 floats  (~331 KB total)

    mpc_prep <<<1, 256, 0, stream>>>(x, gp, ov, Af, Lq, X0, PmT, GT);
    mpc_gemm1<<<4, 256, 0, stream>>>(X0, PmT, PXU);
    mpc_gemm2<<<4, 256, 0, stream>>>(PXU, GT, gp, ov, (float*)d_out);
}